// DNAConv_64707977282142
// MI455X (gfx1250) — compile-verified
//
#include <hip/hip_runtime.h>
#include <math.h>

// Problem structure constants (from the reference): L=3, C=64, H=4, D=16.
#define LHIST 3
#define CCH   64
#define HROW  (LHIST * CCH)   // 192 floats per history row
#define NHEAD 4
#define DHEAD 16

typedef __attribute__((ext_vector_type(2))) float v2f;
typedef __attribute__((ext_vector_type(8))) float v8f;

// ---- monotone float<->uint encoding for atomic max on signed floats ----
__device__ __forceinline__ unsigned enc_f32(float f) {
  unsigned u = __float_as_uint(f);
  return (u & 0x80000000u) ? ~u : (u | 0x80000000u);
}
__device__ __forceinline__ float dec_f32(unsigned u) {
  return (u & 0x80000000u) ? __uint_as_float(u & 0x7FFFFFFFu)
                           : __uint_as_float(~u);
}
#define ENC_NEG_INF 0x007FFFFFu   // enc(-inf)

// ---------------------------------------------------------------------------
// Workspace init: accum=0 (N*64), denom=0 (N*4), segmax=enc(-inf) (N*4)
// ---------------------------------------------------------------------------
__global__ void __launch_bounds__(256)
init_ws_kernel(unsigned* __restrict__ segmax, float* __restrict__ denom,
               float* __restrict__ accum, int n4, int n64) {
  int i = blockIdx.x * blockDim.x + threadIdx.x;
  if (i < n64) accum[i] = 0.0f;
  if (i < n4) { denom[i] = 0.0f; segmax[i] = ENC_NEG_INF; }
}

// ---------------------------------------------------------------------------
// Y[M x 64] = X[M x 64] @ W^T + bias, rows of X at X + r*x_stride + x_off.
// One wave per 16x64 output tile; fp32 V_WMMA_F32_16X16X4_F32, K-loop of 16.
// A layout (16x4 f32): lanes 0-15 hold K=0,1 ; lanes 16-31 hold K=2,3.
// B layout (4x16 f32): lane n holds W[n0+n][k0 + 2*half + {0,1}].
// D layout: vgpr r -> M = r + 8*half, N = lane&15.
// ---------------------------------------------------------------------------
__global__ void __launch_bounds__(256)
gemm64_wmma_kernel(const float* __restrict__ X, int x_stride, int x_off,
                   const float* __restrict__ W, const float* __restrict__ bias,
                   float* __restrict__ Y, int M) {
  int gid  = blockIdx.x * blockDim.x + threadIdx.x;
  int wave = gid >> 5;
  int lane = gid & 31;
  int row0 = wave * 16;
  if (row0 >= M) return;                 // wave-uniform: EXEC stays all-ones
  int half = lane >> 4;
  int ln   = lane & 15;

  const float* __restrict__ xr = X + (size_t)(row0 + ln) * x_stride + x_off;
  v8f acc0 = {}, acc1 = {}, acc2 = {}, acc3 = {};

#pragma unroll
  for (int k0 = 0; k0 < 64; k0 += 4) {
    v2f a;
    a.x = xr[k0 + 2 * half + 0];
    a.y = xr[k0 + 2 * half + 1];
    const float* wp = W + (size_t)ln * 64 + k0 + 2 * half;
    v2f b0, b1, b2, b3;
    b0.x = wp[0];           b0.y = wp[1];
    b1.x = wp[16 * 64 + 0]; b1.y = wp[16 * 64 + 1];
    b2.x = wp[32 * 64 + 0]; b2.y = wp[32 * 64 + 1];
    b3.x = wp[48 * 64 + 0]; b3.y = wp[48 * 64 + 1];
    acc0 = __builtin_amdgcn_wmma_f32_16x16x4_f32(false, a, false, b0, (short)0, acc0, false, false);
    acc1 = __builtin_amdgcn_wmma_f32_16x16x4_f32(false, a, false, b1, (short)0, acc1, false, false);
    acc2 = __builtin_amdgcn_wmma_f32_16x16x4_f32(false, a, false, b2, (short)0, acc2, false, false);
    acc3 = __builtin_amdgcn_wmma_f32_16x16x4_f32(false, a, false, b3, (short)0, acc3, false, false);
  }

  float bb0 = bias[ 0 + ln], bb1 = bias[16 + ln];
  float bb2 = bias[32 + ln], bb3 = bias[48 + ln];
#pragma unroll
  for (int r = 0; r < 8; ++r) {
    int m = row0 + r + 8 * half;
    float* yp = Y + (size_t)m * 64 + ln;
    yp[ 0] = acc0[r] + bb0;
    yp[16] = acc1[r] + bb1;
    yp[32] = acc2[r] + bb2;
    yp[48] = acc3[r] + bb3;
  }
}

// ---------------------------------------------------------------------------
// Edge pass A: scores, per-(dst,head) segment max via encoded atomicMax.
// 16 lanes per edge; butterfly shfl_xor reduction over D=16.
// ---------------------------------------------------------------------------
__global__ void __launch_bounds__(256)
edge_max_kernel(const int* __restrict__ erow, const int* __restrict__ ecol,
                const float* __restrict__ q, const float* __restrict__ k,
                unsigned* __restrict__ segmax, int E) {
  int gid = blockIdx.x * blockDim.x + threadIdx.x;
  int e = gid >> 4;
  if (e >= E) return;                    // uniform per 16-lane group
  int d = gid & 15;
  int r = erow[e], c = ecol[e];
  const float* qp = q + (size_t)c * CCH;
  const float* kp = k + (size_t)r * HROW;
#pragma unroll
  for (int h = 0; h < NHEAD; ++h) {
    float qv = qp[h * DHEAD + d];
    float es = -INFINITY;
#pragma unroll
    for (int l = 0; l < LHIST; ++l) {
      float p = qv * kp[l * CCH + h * DHEAD + d];
      p += __shfl_xor(p, 1);
      p += __shfl_xor(p, 2);
      p += __shfl_xor(p, 4);
      p += __shfl_xor(p, 8);
      es = fmaxf(es, p * 0.25f);         // 1/sqrt(D), D=16
    }
    if (d == 0) atomicMax(segmax + (size_t)c * NHEAD + h, enc_f32(es));
  }
}

// ---------------------------------------------------------------------------
// Edge pass B: recompute scores, token softmax over L, edge weight
// e = exp(max_l score - segmax[col]); accumulate numerator + denominator.
// ---------------------------------------------------------------------------
__global__ void __launch_bounds__(256)
edge_accum_kernel(const int* __restrict__ erow, const int* __restrict__ ecol,
                  const float* __restrict__ q, const float* __restrict__ k,
                  const float* __restrict__ v, const unsigned* __restrict__ segmax,
                  float* __restrict__ denom, float* __restrict__ accum, int E) {
  int gid = blockIdx.x * blockDim.x + threadIdx.x;
  int e = gid >> 4;
  if (e >= E) return;
  int d = gid & 15;
  int r = erow[e], c = ecol[e];
  const float* qp = q + (size_t)c * CCH;
  const float* kp = k + (size_t)r * HROW;
  const float* vp = v + (size_t)r * HROW;
#pragma unroll
  for (int h = 0; h < NHEAD; ++h) {
    float qv = qp[h * DHEAD + d];
    float s[LHIST];
    float es = -INFINITY;
#pragma unroll
    for (int l = 0; l < LHIST; ++l) {
      float p = qv * kp[l * CCH + h * DHEAD + d];
      p += __shfl_xor(p, 1);
      p += __shfl_xor(p, 2);
      p += __shfl_xor(p, 4);
      p += __shfl_xor(p, 8);
      s[l] = p * 0.25f;
      es = fmaxf(es, s[l]);
    }
    // token softmax over L (max is es), message for this lane's channel
    float wsum = 0.0f, msg = 0.0f;
#pragma unroll
    for (int l = 0; l < LHIST; ++l) {
      float w = __expf(s[l] - es);
      wsum += w;
      msg  += w * vp[l * CCH + h * DHEAD + d];
    }
    float sm = dec_f32(segmax[(size_t)c * NHEAD + h]);
    float ew = __expf(es - sm);
    atomicAdd(accum + (size_t)c * CCH + h * DHEAD + d, msg * (ew / wsum));
    if (d == 0) atomicAdd(denom + (size_t)c * NHEAD + h, ew);
  }
}

// ---------------------------------------------------------------------------
// Output projection: out = (accum / denom[head]) @ Wo^T + bo + current.
// Same WMMA tiling as gemm64; A scaled per-head by 1/denom; epilogue fuses
// bias + residual (current = history[:, L-1, :]).
// ---------------------------------------------------------------------------
__global__ void __launch_bounds__(256)
out_proj_wmma_kernel(const float* __restrict__ accum, const float* __restrict__ denom,
                     const float* __restrict__ Wo, const float* __restrict__ bo,
                     const float* __restrict__ history, float* __restrict__ out, int M) {
  int gid  = blockIdx.x * blockDim.x + threadIdx.x;
  int wave = gid >> 5;
  int lane = gid & 31;
  int row0 = wave * 16;
  if (row0 >= M) return;
  int half = lane >> 4;
  int ln   = lane & 15;

  int rA = row0 + ln;
  const float* ar = accum + (size_t)rA * CCH;
  float rd[NHEAD];
#pragma unroll
  for (int h = 0; h < NHEAD; ++h) {
    float dd = denom[(size_t)rA * NHEAD + h];
    rd[h] = dd > 0.0f ? 1.0f / dd : 0.0f;   // no-edge nodes -> 0
  }

  v8f acc0 = {}, acc1 = {}, acc2 = {}, acc3 = {};
#pragma unroll
  for (int k0 = 0; k0 < 64; k0 += 4) {
    float sc = rd[k0 >> 4];                 // k0..k0+3 stay within one head
    v2f a;
    a.x = ar[k0 + 2 * half + 0] * sc;
    a.y = ar[k0 + 2 * half + 1] * sc;
    const float* wp = Wo + (size_t)ln * 64 + k0 + 2 * half;
    v2f b0, b1, b2, b3;
    b0.x = wp[0];           b0.y = wp[1];
    b1.x = wp[16 * 64 + 0]; b1.y = wp[16 * 64 + 1];
    b2.x = wp[32 * 64 + 0]; b2.y = wp[32 * 64 + 1];
    b3.x = wp[48 * 64 + 0]; b3.y = wp[48 * 64 + 1];
    acc0 = __builtin_amdgcn_wmma_f32_16x16x4_f32(false, a, false, b0, (short)0, acc0, false, false);
    acc1 = __builtin_amdgcn_wmma_f32_16x16x4_f32(false, a, false, b1, (short)0, acc1, false, false);
    acc2 = __builtin_amdgcn_wmma_f32_16x16x4_f32(false, a, false, b2, (short)0, acc2, false, false);
    acc3 = __builtin_amdgcn_wmma_f32_16x16x4_f32(false, a, false, b3, (short)0, acc3, false, false);
  }

  float bb0 = bo[ 0 + ln], bb1 = bo[16 + ln];
  float bb2 = bo[32 + ln], bb3 = bo[48 + ln];
#pragma unroll
  for (int r = 0; r < 8; ++r) {
    int m = row0 + r + 8 * half;
    const float* cur = history + (size_t)m * HROW + (LHIST - 1) * CCH;
    float* yp = out + (size_t)m * 64 + ln;
    yp[ 0] = acc0[r] + bb0 + cur[ 0 + ln];
    yp[16] = acc1[r] + bb1 + cur[16 + ln];
    yp[32] = acc2[r] + bb2 + cur[32 + ln];
    yp[48] = acc3[r] + bb3 + cur[48 + ln];
  }
}

// ---------------------------------------------------------------------------
extern "C" void kernel_launch(void* const* d_in, const int* in_sizes, int n_in,
                              void* d_out, int out_size, void* d_ws, size_t ws_size,
                              hipStream_t stream) {
  const float* history = (const float*)d_in[0];
  const int*   edge    = (const int*)d_in[1];
  const float* Wq = (const float*)d_in[2];
  const float* bq = (const float*)d_in[3];
  const float* Wk = (const float*)d_in[4];
  const float* bk = (const float*)d_in[5];
  const float* Wv = (const float*)d_in[6];
  const float* bv = (const float*)d_in[7];
  const float* Wo = (const float*)d_in[8];
  const float* bo = (const float*)d_in[9];
  float* out = (float*)d_out;

  const int N = in_sizes[0] / HROW;       // 50000
  const int E = in_sizes[1] / 2;          // 800000
  const int* erow = edge;                 // edge_index[0] = src
  const int* ecol = edge + E;             // edge_index[1] = dst

  // Workspace layout (fp32): q | k | v | accum | denom | segmax  (~104 MB)
  float* ws    = (float*)d_ws;
  float* qb    = ws;
  float* kb    = qb + (size_t)N * CCH;
  float* vb    = kb + (size_t)N * HROW;
  float* accum = vb + (size_t)N * HROW;
  float* denom = accum + (size_t)N * CCH;
  unsigned* segmax = (unsigned*)(denom + (size_t)N * NHEAD);

  // 0) init accumulators
  {
    int n64 = N * CCH, n4 = N * NHEAD;
    init_ws_kernel<<<(n64 + 255) / 256, 256, 0, stream>>>(segmax, denom, accum, n4, n64);
  }
  // 1) projections (WMMA)
  {
    int Mq = N;
    int thrq = ((Mq + 15) / 16) * 32;
    gemm64_wmma_kernel<<<(thrq + 255) / 256, 256, 0, stream>>>(
        history, HROW, (LHIST - 1) * CCH, Wq, bq, qb, Mq);
    int Mkv = N * LHIST;
    int thrkv = ((Mkv + 15) / 16) * 32;
    gemm64_wmma_kernel<<<(thrkv + 255) / 256, 256, 0, stream>>>(
        history, CCH, 0, Wk, bk, kb, Mkv);
    gemm64_wmma_kernel<<<(thrkv + 255) / 256, 256, 0, stream>>>(
        history, CCH, 0, Wv, bv, vb, Mkv);
  }
  // 2) edge phase: segment max, then weighted accumulation
  {
    long threads = (long)E * 16;
    int blocks = (int)((threads + 255) / 256);
    edge_max_kernel<<<blocks, 256, 0, stream>>>(erow, ecol, qb, kb, segmax, E);
    edge_accum_kernel<<<blocks, 256, 0, stream>>>(erow, ecol, qb, kb, vb, segmax,
                                                  denom, accum, E);
  }
  // 3) output projection + residual (WMMA)
  {
    int thr = ((N + 15) / 16) * 32;
    out_proj_wmma_kernel<<<(thr + 255) / 256, 256, 0, stream>>>(
        accum, denom, Wo, bo, history, out, N);
  }
}